// QueryPooling_18485539242901
// MI455X (gfx1250) — compile-verified
//
#include <hip/hip_runtime.h>
#include <float.h>

// CDNA5 / gfx1250, wave32. Fused flash-style query-pooling:
// one HBM pass over x (pass B re-reads each 512KB chunk from L2), f32 WMMA 16x16x4.

typedef __attribute__((ext_vector_type(2))) float v2f;
typedef __attribute__((ext_vector_type(8))) float v8f;

#define B_ 16
#define L_ 4096
#define C_ 1024
#define K_ 16
#define NSPLIT 16                 // L-splits per batch -> 256 workgroups
#define LRANGE (L_ / NSPLIT)      // 256 rows per workgroup
#define CHUNK 128                 // rows per online-softmax chunk (512KB of x, L2-resident)
#define NCHUNK (LRANGE / CHUNK)   // 2
#define SP 132                    // LDS row stride: 132 % 64 == 4 -> conflict-free b64 frag reads
#define CW (C_ / 8)               // 128 columns per wave in pass B
#define NT 8                      // 16-wide c-tiles per wave
#define NEG_MAX (-3.402823466e38f)

// ---------------------------------------------------------------------------
// Fused kernel: per (b, L-split): scores -> online softmax -> pooled partial.
// WMMA fragment layouts (ISA 7.12.2, f32 16x16x4):
//   A (16x4, MxK): lane = m + 16*(k>>1), reg = k&1
//   B (4x16, KxN): lane = n + 16*(k>>1), reg = k&1
//   C/D (16x16):   vgpr r: lanes0-15 (M=r, N=lane), lanes16-31 (M=r+8, N=lane-16)
// ---------------------------------------------------------------------------
__global__ __launch_bounds__(256) void qp_fused(const float* __restrict__ x,
                                                const int* __restrict__ mask,
                                                const float* __restrict__ qm,
                                                float* __restrict__ part,
                                                float* __restrict__ mstat,
                                                float* __restrict__ sstat)
{
    __shared__ float sw[K_ * SP];     // chunk scores -> exp-weights, [16][SP]
    __shared__ float red[K_ * 16];    // block reduction scratch
    __shared__ float mq[K_], sq[K_], scq[K_];   // running max / sum / rescale

    const int tid   = threadIdx.x;
    const int w     = tid >> 5;       // wave 0..7
    const int lane  = tid & 31;
    const int lid   = lane & 15;
    const int half  = lane >> 4;
    const int split = blockIdx.x & (NSPLIT - 1);
    const int b     = blockIdx.x >> 4;
    const int lbase = split * LRANGE;
    const int cw    = w * CW;         // this wave's column window in pass B

    const int tq = tid >> 4;          // softmax mapping: thread -> (query, lane-in-16)
    const int tj = tid & 15;

    if (tid < K_) { mq[tid] = NEG_MAX; sq[tid] = 0.0f; }
    __syncthreads();

    v8f zero = {};
    v8f acc[NT];
#pragma unroll
    for (int j = 0; j < NT; ++j) acc[j] = zero;

    const float* qrow = qm + lid * C_ + 2 * half;

    for (int ch = 0; ch < NCHUNK; ++ch) {
        // ---------------- pass A: score tile 16q x 16l per wave ------------
        // x loads stay RT: they seed L2 for the pass-B re-read.
        // unroll 16 -> ~32 b64 loads in flight/wave (latency-BW product at
        // 2 waves/SIMD needs ~55KB in flight per WGP to hold 23.3 TB/s).
        const int l0 = lbase + ch * CHUNK + w * 16;
        const float* xrow = x + (size_t)(b * L_ + l0 + lid) * C_ + 2 * half;
        v8f d = zero;
#pragma unroll 16
        for (int c = 0; c < C_; c += 4) {
            v2f av = *(const v2f*)(qrow + c);
            v2f bv = *(const v2f*)(xrow + c);
            d = __builtin_amdgcn_wmma_f32_16x16x4_f32(false, av, false, bv,
                                                      (short)0, d, false, false);
        }
        const int mval = mask[b * L_ + l0 + lid];
        const int lloc = w * 16 + lid;
#pragma unroll
        for (int r = 0; r < 8; ++r) {
            float v = d[r] * 0.03125f;                 // C^-0.5 == 1/32 exactly
            sw[(r + 8 * half) * SP + lloc] = mval ? v : NEG_MAX;
        }
        __syncthreads();

        // ---------------- online softmax update ----------------------------
        float lm = NEG_MAX;
#pragma unroll
        for (int i = 0; i < CHUNK / 16; ++i)
            lm = fmaxf(lm, sw[tq * SP + tj + 16 * i]);
        red[tq * 16 + tj] = lm;
        __syncthreads();
        if (tid < K_) {
            float nm = mq[tid];
            for (int i = 0; i < 16; ++i) nm = fmaxf(nm, red[tid * 16 + i]);
            scq[tid] = __expf(mq[tid] - nm);  // 0 when first finite max appears
            mq[tid]  = nm;
        }
        __syncthreads();

        // rescale pooled accumulators by per-query factor (q = r + 8*half)
        float scl[8];
#pragma unroll
        for (int r = 0; r < 8; ++r) scl[r] = scq[r + 8 * half];
#pragma unroll
        for (int j = 0; j < NT; ++j)
#pragma unroll
            for (int r = 0; r < 8; ++r) acc[j][r] *= scl[r];

        // exponentiate chunk in LDS (scores -> weights), accumulate row sums
        const float nmv = mq[tq];
        float ls = 0.0f;
#pragma unroll
        for (int i = 0; i < CHUNK / 16; ++i) {
            float e = __expf(sw[tq * SP + tj + 16 * i] - nmv);
            sw[tq * SP + tj + 16 * i] = e;
            ls += e;
        }
        red[tq * 16 + tj] = ls;
        __syncthreads();
        if (tid < K_) {
            float s = 0.0f;
            for (int i = 0; i < 16; ++i) s += red[tid * 16 + i];
            sq[tid] = sq[tid] * scq[tid] + s;
        }

        // ---------------- pass B: acc += weights(16xCHUNK) . x(CHUNK x CW) --
        // x chunk re-read: 256 WGs x 512KB = 128MB working set -> L2-served.
        // NT hint: these lines are dead after this read; don't hold L2.
        const float* xcol = x + (size_t)(b * L_ + lbase + ch * CHUNK) * C_ + cw + lid;
        for (int l = 0; l < CHUNK; l += 4) {
            v2f av = *(const v2f*)(&sw[lid * SP + l + 2 * half]);  // conflict-free b64
            const float* xp = xcol + (size_t)(l + 2 * half) * C_;
#pragma unroll
            for (int j = 0; j < NT; ++j) {
                v2f bv;
                bv.x = __builtin_nontemporal_load(xp + j * 16);
                bv.y = __builtin_nontemporal_load(xp + C_ + j * 16);
                acc[j] = __builtin_amdgcn_wmma_f32_16x16x4_f32(false, av, false, bv,
                                                               (short)0, acc[j], false, false);
            }
        }
        __syncthreads();   // sw free for next chunk's pass A
    }

    // ---------------- epilogue: partials + stats (write-once -> NT) --------
    float* prow = part + ((size_t)((b * NSPLIT + split) * K_ + 8 * half)) * C_ + cw + lid;
#pragma unroll
    for (int j = 0; j < NT; ++j)
#pragma unroll
        for (int r = 0; r < 8; ++r)
            __builtin_nontemporal_store(acc[j][r], prow + (size_t)r * C_ + j * 16);

    if (tid < K_) {
        mstat[(b * NSPLIT + split) * K_ + tid] = mq[tid];
        sstat[(b * NSPLIT + split) * K_ + tid] = sq[tid];
    }
}

// ---------------------------------------------------------------------------
// Combine: flash-decode merge of NSPLIT partials per (b,q,c).
// out = sum_w exp(m_w - M) * acc_w / sum_w exp(m_w - M) * s_w ; 0 if no valid token.
// ---------------------------------------------------------------------------
__global__ __launch_bounds__(256) void qp_combine(const float* __restrict__ part,
                                                  const float* __restrict__ mstat,
                                                  const float* __restrict__ sstat,
                                                  float* __restrict__ out)
{
    const int i = blockIdx.x * 256 + threadIdx.x;   // (b*K + q)*C + c
    const int c = i & (C_ - 1);
    const int q = (i >> 10) & (K_ - 1);
    const int b = i >> 14;

    float M = NEG_MAX;
#pragma unroll
    for (int wv = 0; wv < NSPLIT; ++wv)
        M = fmaxf(M, mstat[(b * NSPLIT + wv) * K_ + q]);

    if (M == NEG_MAX) { out[i] = 0.0f; return; }    // no valid token in batch row

    float num = 0.0f, den = 0.0f;
#pragma unroll
    for (int wv = 0; wv < NSPLIT; ++wv) {
        const int sidx = (b * NSPLIT + wv) * K_ + q;
        const float e  = __expf(mstat[sidx] - M);
        den += e * sstat[sidx];
        num += e * __builtin_nontemporal_load(part + (size_t)sidx * C_ + c);
    }
    out[i] = num / den;
}

extern "C" void kernel_launch(void* const* d_in, const int* in_sizes, int n_in,
                              void* d_out, int out_size, void* d_ws, size_t ws_size,
                              hipStream_t stream)
{
    const float* x    = (const float*)d_in[0];   // [B, L, C] f32
    const int*   mask = (const int*)d_in[1];     // [B, L] i32
    const float* q    = (const float*)d_in[2];   // [K, C] f32
    float*       out  = (float*)d_out;           // [B, K, C] f32

    float* part  = (float*)d_ws;                                   // 16 MB
    float* mstat = part + (size_t)B_ * NSPLIT * K_ * C_;           // 16 KB
    float* sstat = mstat + B_ * NSPLIT * K_;                       // 16 KB

    qp_fused<<<B_ * NSPLIT, 256, 0, stream>>>(x, mask, q, part, mstat, sstat);
    qp_combine<<<(B_ * K_ * C_) / 256, 256, 0, stream>>>(part, mstat, sstat, out);
}